// GroupedVQ_4672924418780
// MI455X (gfx1250) — compile-verified
//
#include <hip/hip_runtime.h>
#include <hip/hip_bf16.h>

typedef __attribute__((ext_vector_type(2))) float v2f;
typedef __attribute__((ext_vector_type(8))) float v8f;

#define G_    4
#define C_    2048
#define DG    128
#define NTOT  16384     // B*S = 8*2048
#define D_    512
#define DECAY 0.8f
#define EPSL  1e-5f
#define COMMIT 1.0f

// ---------------------------------------------------------------- e-norms
__global__ __launch_bounds__(128) void k_enorm(const float* __restrict__ embed,
                                               float* __restrict__ e_norms) {
    __shared__ float red[128];
    int row = blockIdx.x;            // 0 .. G_*C_-1
    int t = threadIdx.x;             // 0..127
    float v = embed[(size_t)row * DG + t];
    red[t] = v * v;
    __syncthreads();
    for (int s = 64; s > 0; s >>= 1) {
        if (t < s) red[t] += red[t + s];
        __syncthreads();
    }
    if (t == 0) e_norms[row] = red[0];
}

// ------------------------------------------- distance GEMM (WMMA) + argmax
// block: 256 threads = 8 wave32; block owns 128 rows (n) of one group g.
// Each wave computes TWO 16-column tiles per step (independent WMMA chains
// -> partial dscnt waits, LDS latency hidden), double-buffered LDS staging.
__global__ __launch_bounds__(256) void k_dist_argmax(
        const float* __restrict__ x, const float* __restrict__ embed,
        const float* __restrict__ e_norms, int* __restrict__ ind_ws) {
    __shared__ float lds_e[2][32 * 132];       // 32 codes x 128 k, pad->132
    const int g    = blockIdx.x >> 7;          // 128 n-blocks per group
    const int n0   = (blockIdx.x & 127) * 128;
    const int t    = threadIdx.x;
    const int wave = t >> 5;
    const int lane = t & 31;
    const int hi   = lane >> 4;                // selects K+2,K+3 half
    const int lo   = lane & 15;

    const float* ebase = embed + (size_t)g * C_ * DG;

    // A fragments for this wave's 16 rows: lane lo = row, hi half = K+2/K+3
    const int row = n0 + wave * 16 + lo;
    const float* xrow = x + (size_t)row * D_ + g * DG;
    v2f a[32];
#pragma unroll
    for (int kk = 0; kk < 32; ++kk)
        a[kk] = *(const v2f*)(xrow + kk * 4 + hi * 2);

    float bestv[8];
    int   besti[8];
#pragma unroll
    for (int r = 0; r < 8; ++r) { bestv[r] = -3.4e38f; besti[r] = 0; }

    // staging: 32 codes x 128 floats = 1024 float4's, 4 per thread
    float4 pre[4];
#pragma unroll
    for (int j = 0; j < 4; ++j) {
        int idx = t + j * 256;
        int cc = idx >> 5, k4 = (idx & 31) * 4;
        pre[j] = *(const float4*)(ebase + (size_t)cc * DG + k4);
    }

    int buf = 0;
    for (int c0 = 0; c0 < C_; c0 += 32) {
        // commit prefetched tile to LDS
#pragma unroll
        for (int j = 0; j < 4; ++j) {
            int idx = t + j * 256;
            int cc = idx >> 5, k4 = (idx & 31) * 4;
            *(float4*)(&lds_e[buf][cc * 132 + k4]) = pre[j];
        }
        __syncthreads();

        // issue global prefetch for the NEXT tile (overlaps WMMA chains)
        if (c0 + 32 < C_) {
            const float* nb = ebase + (size_t)(c0 + 32) * DG;
#pragma unroll
            for (int j = 0; j < 4; ++j) {
                int idx = t + j * 256;
                int cc = idx >> 5, k4 = (idx & 31) * 4;
                pre[j] = *(const float4*)(nb + (size_t)cc * DG + k4);
            }
        }

        // two independent WMMA chains over the two 16-code sub-tiles
        const float* browA = &lds_e[buf][lo * 132 + hi * 2];
        const float* browB = browA + 16 * 132;
        v8f accA = {0.f, 0.f, 0.f, 0.f, 0.f, 0.f, 0.f, 0.f};
        v8f accB = {0.f, 0.f, 0.f, 0.f, 0.f, 0.f, 0.f, 0.f};
#pragma unroll
        for (int kk = 0; kk < 32; ++kk) {
            v2f bA = *(const v2f*)(browA + kk * 4);
            v2f bB = *(const v2f*)(browB + kk * 4);
            accA = __builtin_amdgcn_wmma_f32_16x16x4_f32(
                       false, a[kk], false, bA, (short)0, accA, false, false);
            accB = __builtin_amdgcn_wmma_f32_16x16x4_f32(
                       false, a[kk], false, bB, (short)0, accB, false, false);
        }

        const int colA = c0 + lo;              // tile-A column for this lane
        const int colB = colA + 16;            // tile-B column
        const float enA = e_norms[g * C_ + colA];
        const float enB = e_norms[g * C_ + colB];
#pragma unroll
        for (int r = 0; r < 8; ++r) {
            float sA = 2.0f * accA[r] - enA;   // dist + const(||f||^2)
            if (sA > bestv[r]) { bestv[r] = sA; besti[r] = colA; }
            float sB = 2.0f * accB[r] - enB;   // colB > colA: strict > keeps ties
            if (sB > bestv[r]) { bestv[r] = sB; besti[r] = colB; }
        }
        buf ^= 1;
    }

    // argmax across the 16 lanes of each half (rows r / r+8), first-index ties
#pragma unroll
    for (int off = 1; off < 16; off <<= 1) {
#pragma unroll
        for (int r = 0; r < 8; ++r) {
            float ov = __shfl_xor(bestv[r], off, 32);
            int   oi = __shfl_xor(besti[r], off, 32);
            if (ov > bestv[r] || (ov == bestv[r] && oi < besti[r])) {
                bestv[r] = ov; besti[r] = oi;
            }
        }
    }
    if (lo == 0) {
#pragma unroll
        for (int r = 0; r < 8; ++r) {
            int m = r + hi * 8;                // C/D layout: half -> +8 rows
            ind_ws[g * NTOT + n0 + wave * 16 + m] = besti[r];
        }
    }
}

// -------------------------------- gather quantize, ind->float, loss partials
__global__ __launch_bounds__(256) void k_gather_loss(
        const float* __restrict__ x, const float* __restrict__ embed,
        const int* __restrict__ ind_ws, float* __restrict__ out_q,
        float* __restrict__ out_ind, float* __restrict__ partials) {
    __shared__ float red[256];
    __shared__ int cidx[G_];
    int n = blockIdx.x;
    int t = threadIdx.x;
    if (t < G_) {
        int c = ind_ws[t * NTOT + n];
        cidx[t] = c;
        out_ind[(size_t)n * G_ + t] = (float)c;
    }
    __syncthreads();
    float sq = 0.f;
#pragma unroll
    for (int j = 0; j < 2; ++j) {
        int ch = t + j * 256;
        int g = ch >> 7, d = ch & 127;
        float q  = embed[((size_t)(g * C_ + cidx[g])) * DG + d];
        float xv = x[(size_t)n * D_ + ch];
        out_q[(size_t)n * D_ + ch] = q;        // quantize_st == quantize
        float dd = q - xv;
        sq += dd * dd;
    }
    red[t] = sq;
    __syncthreads();
    for (int s = 128; s > 0; s >>= 1) {
        if (t < s) red[t] += red[t + s];
        __syncthreads();
    }
    if (t == 0) partials[n] = red[0];
}

__global__ __launch_bounds__(256) void k_loss_reduce(
        const float* __restrict__ partials, float* __restrict__ out_loss) {
    __shared__ float red[256];
    int t = threadIdx.x;
    float s = 0.f;
    for (int i = t; i < NTOT; i += 256) s += partials[i];
    red[t] = s;
    __syncthreads();
    for (int k = 128; k > 0; k >>= 1) {
        if (t < k) red[t] += red[t + k];
        __syncthreads();
    }
    if (t == 0) out_loss[0] = (COMMIT * red[0]) / (float)(NTOT * D_);
}

// ------------- bins + segment_sum + EMA (one block owns one (g,c): no atomics)
__global__ __launch_bounds__(128) void k_binsum(
        const float* __restrict__ x, const int* __restrict__ ind_ws,
        const float* __restrict__ cluster_size,
        const float* __restrict__ embed_avg,
        float* __restrict__ out_ncs, float* __restrict__ out_nea) {
    __shared__ int lind[128];
    int gc = blockIdx.x;
    int g = gc >> 11;                          // /C_
    int c = gc & (C_ - 1);
    int t = threadIdx.x;
    float acc = 0.f;
    int cnt = 0;
    for (int nb = 0; nb < NTOT; nb += 128) {
        __syncthreads();
        lind[t] = ind_ws[g * NTOT + nb + t];
        __syncthreads();
        for (int j = 0; j < 128; ++j) {        // uniform branch across block
            if (lind[j] == c) {
                acc += x[(size_t)(nb + j) * D_ + g * DG + t];
                ++cnt;
            }
        }
    }
    out_nea[(size_t)gc * DG + t] =
        embed_avg[(size_t)gc * DG + t] * DECAY + (1.f - DECAY) * acc;
    if (t == 0)
        out_ncs[gc] = cluster_size[gc] * DECAY + (1.f - DECAY) * (float)cnt;
}

__global__ __launch_bounds__(256) void k_totals(const float* __restrict__ out_ncs,
                                                float* __restrict__ totals) {
    __shared__ float red[256];
    int g = blockIdx.x, t = threadIdx.x;
    float s = 0.f;
    for (int c = t; c < C_; c += 256) s += out_ncs[g * C_ + c];
    red[t] = s;
    __syncthreads();
    for (int k = 128; k > 0; k >>= 1) {
        if (t < k) red[t] += red[t + k];
        __syncthreads();
    }
    if (t == 0) totals[g] = red[0];
}

__global__ __launch_bounds__(128) void k_newembed(
        const float* __restrict__ out_ncs, const float* __restrict__ out_nea,
        const float* __restrict__ totals, float* __restrict__ out_e) {
    int gc = blockIdx.x;
    int g = gc >> 11;
    int t = threadIdx.x;
    float total = totals[g];
    float ncs = out_ncs[gc];
    float smoothed = (ncs + EPSL) / (total + (float)C_ * EPSL) * total;
    out_e[(size_t)gc * DG + t] = out_nea[(size_t)gc * DG + t] / smoothed;
}

// ---------------------------------------------------------------- launcher
extern "C" void kernel_launch(void* const* d_in, const int* in_sizes, int n_in,
                              void* d_out, int out_size, void* d_ws, size_t ws_size,
                              hipStream_t stream) {
    const float* x            = (const float*)d_in[0];   // [8,2048,512]
    const float* embed        = (const float*)d_in[1];   // [4,2048,128]
    const float* cluster_size = (const float*)d_in[2];   // [4,2048]
    const float* embed_avg    = (const float*)d_in[3];   // [4,2048,128]

    float* out = (float*)d_out;
    // output regions, flat in return order
    float* out_q    = out;                               // 8,388,608
    float* out_ind  = out + (size_t)NTOT * D_;           //    65,536
    float* out_loss = out_ind + (size_t)NTOT * G_;       //         1
    float* out_ncs  = out_loss + 1;                      //     8,192
    float* out_nea  = out_ncs + (size_t)G_ * C_;         // 1,048,576
    float* out_e    = out_nea + (size_t)G_ * C_ * DG;    // 1,048,576

    // workspace layout
    float* wsf      = (float*)d_ws;
    float* e_norms  = wsf;                               //  8,192 f
    int*   ind_ws   = (int*)(wsf + G_ * C_);             // 65,536 i
    float* partials = wsf + G_ * C_ + G_ * NTOT;         // 16,384 f
    float* totals   = partials + NTOT;                   //      4 f

    k_enorm      <<<G_ * C_,        128, 0, stream>>>(embed, e_norms);
    k_dist_argmax<<<G_ * (NTOT/128),256, 0, stream>>>(x, embed, e_norms, ind_ws);
    k_gather_loss<<<NTOT,           256, 0, stream>>>(x, embed, ind_ws,
                                                      out_q, out_ind, partials);
    k_loss_reduce<<<1,              256, 0, stream>>>(partials, out_loss);
    k_binsum     <<<G_ * C_,        128, 0, stream>>>(x, ind_ws, cluster_size,
                                                      embed_avg, out_ncs, out_nea);
    k_totals     <<<G_,             256, 0, stream>>>(out_ncs, totals);
    k_newembed   <<<G_ * C_,        128, 0, stream>>>(out_ncs, out_nea, totals, out_e);
}